// Get_gradient_nopadding_14886356648203
// MI455X (gfx1250) — compile-verified
//
#include <hip/hip_runtime.h>
#include <stdint.h>

// Sobel gradient magnitude, 96 independent 512x512 fp32 planes.
// Bandwidth-bound: stage tile+halo rows into LDS with CDNA5 async
// global->LDS copies (ASYNCcnt), compute from LDS, NT-store the output.

typedef float v4f __attribute__((ext_vector_type(4)));

#define IMG_W 512
#define IMG_H 512
#define TILE_H 16
#define NTHREADS 256
#define HALO_H (TILE_H + 2)

__global__ __launch_bounds__(NTHREADS)
void sobel_mag_kernel(const float* __restrict__ x, float* __restrict__ out) {
    __shared__ float tile[HALO_H * IMG_W];   // 18 * 512 * 4B = 36 KB

    const int tid   = threadIdx.x;
    const int tIdx  = blockIdx.x;            // tile row-band within plane
    const int plane = blockIdx.y;            // batch*channel plane
    const int row0  = tIdx * TILE_H;

    const float* xp = x   + (size_t)plane * (IMG_W * IMG_H);
    float*       op = out + (size_t)plane * (IMG_W * IMG_H);
    const uint64_t xbase = (uint64_t)(uintptr_t)xp;

    // ---- Stage rows [row0-1, row0+TILE_H] into LDS -------------------------
    // 18 rows * 512 floats = 9216 floats; 256 threads * 4 floats = 1024/pass
    // -> 9 passes. Row index changes only at 128-thread boundaries, so the
    // in-range predicate is uniform per wave32 (async loads issue with full
    // EXEC). Out-of-range halo rows (zero padding) are zero-filled in LDS.
    for (int base = tid * 4; base < HALO_H * IMG_W; base += NTHREADS * 4) {
        const int r  = base >> 9;            // base / 512
        const int gr = row0 - 1 + r;         // global row, may be -1 or 512
        // LDS byte offset: low 32 bits of the flat pointer to the LDS array.
        const uint32_t lds_addr = (uint32_t)(uintptr_t)(&tile[base]);
        if ((unsigned)gr < (unsigned)IMG_H) {
            const int c = base & (IMG_W - 1);
            const uint32_t goff = (uint32_t)(((gr << 9) + c) << 2);
            // GVS form: vdst = LDS byte addr, vaddr = 32-bit offset, saddr = base.
            asm volatile("global_load_async_to_lds_b128 %0, %1, %2"
                         :
                         : "v"(lds_addr), "v"(goff), "s"(xbase)
                         : "memory");
        } else {
            v4f z = {0.f, 0.f, 0.f, 0.f};
            *(v4f*)&tile[base] = z;          // zero padding rows
        }
    }
    // Wait for this wave's async copies; barrier makes all waves' data visible.
    asm volatile("s_wait_asynccnt 0" ::: "memory");
    __syncthreads();

    // ---- Compute 4 outputs per thread per row ------------------------------
    // Threads 0..127 -> even rows of the pair, 128..255 -> odd rows.
    const int c      = (tid & (IMG_W / 4 - 1)) * 4;  // 0..508, step 4
    const int rstart = tid >> 7;                     // 0 or 1
    for (int rr = rstart; rr < TILE_H; rr += 2) {
        const v4f t = *(const v4f*)&tile[(rr    ) * IMG_W + c];  // row gr-1
        const v4f m = *(const v4f*)&tile[(rr + 1) * IMG_W + c];  // row gr
        const v4f b = *(const v4f*)&tile[(rr + 2) * IMG_W + c];  // row gr+1

        const float left  = (c > 0)          ? tile[(rr + 1) * IMG_W + c - 1] : 0.f;
        const float right = (c + 4 < IMG_W)  ? tile[(rr + 1) * IMG_W + c + 4] : 0.f;

        v4f gv = b - t;                       // vertical 2-tap
        v4f gh;                               // horizontal 2-tap (zero pad edges)
        gh.x = m.y - left;
        gh.y = m.z - m.x;
        gh.z = m.w - m.y;
        gh.w = right - m.z;

        v4f s = gv * gv + gh * gh + 1e-6f;
        v4f o;
        o.x = sqrtf(s.x);
        o.y = sqrtf(s.y);
        o.z = sqrtf(s.z);
        o.w = sqrtf(s.w);

        // Output is write-once / never re-read: non-temporal b128 store keeps
        // L2 free for the input stream.
        __builtin_nontemporal_store(o, (v4f*)(op + (row0 + rr) * IMG_W + c));
    }
}

extern "C" void kernel_launch(void* const* d_in, const int* in_sizes, int n_in,
                              void* d_out, int out_size, void* d_ws, size_t ws_size,
                              hipStream_t stream) {
    const float* x = (const float*)d_in[0];
    float* out = (float*)d_out;

    const int planes = in_sizes[0] / (IMG_W * IMG_H);   // 32 * 3 = 96
    dim3 grid(IMG_H / TILE_H, planes);                  // (32, 96)
    sobel_mag_kernel<<<grid, NTHREADS, 0, stream>>>(x, out);
}